// GCN_PyG_11175504904536
// MI455X (gfx1250) — compile-verified
//
#include <hip/hip_runtime.h>
#include <hip/hip_bf16.h>

typedef __attribute__((ext_vector_type(2))) float v2f;
typedef __attribute__((ext_vector_type(8))) float v8f;

#define KDIM 128

// ---------------- degree / normalization ----------------

__global__ __launch_bounds__(256) void k_init_deg(float* deg, int n) {
    int i = blockIdx.x * blockDim.x + threadIdx.x;
    if (i < n) deg[i] = 1.0f;  // self-loop contributes 1
}

__global__ __launch_bounds__(256) void k_deg_scatter(const long long* __restrict__ dst,
                                                     float* __restrict__ deg, int e) {
    int i = blockIdx.x * blockDim.x + threadIdx.x;
    if (i < e) atomicAdd(&deg[dst[i]], 1.0f);
}

__global__ __launch_bounds__(256) void k_rsqrt(float* deg, int n) {
    int i = blockIdx.x * blockDim.x + threadIdx.x;
    if (i < n) deg[i] = rsqrtf(deg[i]);  // deg >= 1 always (self-loop)
}

__global__ __launch_bounds__(256) void k_zero4(float4* p, int n4) {
    int i = blockIdx.x * blockDim.x + threadIdx.x;
    if (i < n4) p[i] = make_float4(0.f, 0.f, 0.f, 0.f);
}

// ---------------- WMMA GEMM ----------------
// C[M,N] = (A (+A2)) @ W[K=128,N], optional row-scale (pre-aggregation dis) & bias.
// One wave computes a 16x64 tile: 4 accumulators share one A fragment per k-step
// (4x less A traffic, 4 independent back-to-back v_wmma_f32_16x16x4_f32).
// A 16x4 layout: lanes 0-15 -> K=k0+0,k0+1 ; lanes 16-31 -> K=k0+2,k0+3 ; row = lane&15.
// B 4x16 symmetric (col = lane&15). C: VGPR v -> M = v + 8*(lane>=16), N = lane&15.

template <bool FUSE_ADD, bool HAS_SCALE, bool HAS_BIAS>
__global__ __launch_bounds__(128) void k_gemm_wmma(
    const float* __restrict__ A, const float* __restrict__ A2,
    const float* __restrict__ W, const float* __restrict__ bias,
    const float* __restrict__ rowscale, float* __restrict__ C, int M, int N) {
    const int wave = threadIdx.x >> 5;
    const int lane = threadIdx.x & 31;
    const int tileM = blockIdx.x * 4 + wave;
    if (tileM * 16 >= M) return;  // wave-uniform: EXEC stays all-ones for WMMA

    const int nBase = blockIdx.y * 64;  // this wave covers cols [nBase, nBase+64)
    const int half = lane >> 4;         // 0 or 1
    const int l    = lane & 15;
    const int koff = half * 2;          // K sub-offset handled by lane half

    const long long rowA = (long long)tileM * 16 + l;
    const float* __restrict__ Ap  = A + rowA * KDIM + koff;
    const float* __restrict__ A2p = FUSE_ADD ? (A2 + rowA * KDIM + koff) : nullptr;
    const float* __restrict__ Wp  = W + (long long)koff * N + nBase + l;

    v8f acc0 = {0.f, 0.f, 0.f, 0.f, 0.f, 0.f, 0.f, 0.f};
    v8f acc1 = acc0, acc2 = acc0, acc3 = acc0;

#pragma unroll 4
    for (int k0 = 0; k0 < KDIM; k0 += 4) {
        v2f a;
        a.x = Ap[k0];
        a.y = Ap[k0 + 1];
        if (FUSE_ADD) {
            a.x += A2p[k0];
            a.y += A2p[k0 + 1];
        }
        const float* __restrict__ w0 = Wp + (long long)k0 * N;  // K = k0+koff
        const float* __restrict__ w1 = w0 + N;                  // K = k0+koff+1
        v2f b0, b1, b2, b3;
        b0.x = w0[0];  b0.y = w1[0];
        b1.x = w0[16]; b1.y = w1[16];
        b2.x = w0[32]; b2.y = w1[32];
        b3.x = w0[48]; b3.y = w1[48];
        acc0 = __builtin_amdgcn_wmma_f32_16x16x4_f32(false, a, false, b0, (short)0, acc0, false, false);
        acc1 = __builtin_amdgcn_wmma_f32_16x16x4_f32(false, a, false, b1, (short)0, acc1, false, false);
        acc2 = __builtin_amdgcn_wmma_f32_16x16x4_f32(false, a, false, b2, (short)0, acc2, false, false);
        acc3 = __builtin_amdgcn_wmma_f32_16x16x4_f32(false, a, false, b3, (short)0, acc3, false, false);
    }

    // ---- epilogue: rows tileM*16 + half*8 + v, v = 0..7 ----
    float scale[8];
    if (HAS_SCALE) {
        const float* rs = rowscale + (long long)tileM * 16 + half * 8;
#pragma unroll
        for (int v = 0; v < 8; v++) scale[v] = rs[v];
    }
    const long long rowBase = (long long)tileM * 16 + half * 8;
#pragma unroll
    for (int j = 0; j < 4; j++) {
        const v8f& acc = (j == 0) ? acc0 : (j == 1) ? acc1 : (j == 2) ? acc2 : acc3;
        const int col = nBase + j * 16 + l;
        const float bj = HAS_BIAS ? bias[col] : 0.f;
        float* __restrict__ Cp = C + rowBase * N + col;
#pragma unroll
        for (int v = 0; v < 8; v++) {
            float val = acc[v];
            if (HAS_SCALE) val *= scale[v];
            if (HAS_BIAS) val += bj;
            Cp[(long long)v * N] = val;
        }
    }
}

// ---------------- edge scatter: agg[dst] += h'[src] (one wave per edge, float4/lane) ----------------

__global__ __launch_bounds__(256) void k_scatter(const long long* __restrict__ src,
                                                 const long long* __restrict__ dst,
                                                 const float* __restrict__ h,
                                                 float* __restrict__ agg, int e) {
    int t = blockIdx.x * blockDim.x + threadIdx.x;
    int edge = t >> 5;
    int lane = t & 31;
    if (edge >= e) return;
    long long s = src[edge];
    long long d = dst[edge];
    const float4 v = *(const float4*)(h + s * KDIM + lane * 4);
    float* o = agg + d * KDIM + lane * 4;
    atomicAdd(o + 0, v.x);
    atomicAdd(o + 1, v.y);
    atomicAdd(o + 2, v.z);
    atomicAdd(o + 3, v.w);
}

// ---------------- layer epilogue: x = relu(dis*(agg + h') + b) ----------------

__global__ __launch_bounds__(256) void k_epilogue(const float* __restrict__ agg,
                                                  const float* __restrict__ h,
                                                  const float* __restrict__ dis,
                                                  const float* __restrict__ b,
                                                  float* __restrict__ x, int n) {
    int t = blockIdx.x * blockDim.x + threadIdx.x;
    int row = t >> 5;
    int c0 = (t & 31) * 4;
    if (row >= n) return;
    float s = dis[row];
    long long base = (long long)row * KDIM + c0;
    float4 a  = *(const float4*)(agg + base);
    float4 hh = *(const float4*)(h + base);
    float4 bb = *(const float4*)(b + c0);
    float4 r;
    r.x = fmaxf(s * (a.x + hh.x) + bb.x, 0.f);
    r.y = fmaxf(s * (a.y + hh.y) + bb.y, 0.f);
    r.z = fmaxf(s * (a.z + hh.z) + bb.z, 0.f);
    r.w = fmaxf(s * (a.w + hh.w) + bb.w, 0.f);
    *(float4*)(x + base) = r;
}

// ---------------- launch ----------------

extern "C" void kernel_launch(void* const* d_in, const int* in_sizes, int n_in,
                              void* d_out, int out_size, void* d_ws, size_t ws_size,
                              hipStream_t stream) {
    const float* x  = (const float*)d_in[0];
    const long long* ei = (const long long*)d_in[1];  // int64 edge_index [2, E]
    const float* W1 = (const float*)d_in[2];
    const float* b1 = (const float*)d_in[3];
    const float* W2 = (const float*)d_in[4];
    const float* b2 = (const float*)d_in[5];
    const float* Wo = (const float*)d_in[6];
    const float* bo = (const float*)d_in[7];

    const int N = in_sizes[0] / KDIM;   // 100000 nodes
    const int E = in_sizes[1] / 2;      // 1.6M edges
    const int NOUT = in_sizes[7];       // 64

    const long long* src = ei;
    const long long* dst = ei + E;

    // workspace layout (floats)
    float* ws = (float*)d_ws;
    const long long nPad = ((long long)N + 1023) & ~1023LL;
    const long long feat = (long long)N * KDIM;
    float* dis  = ws;                 // [N]  (deg, then rsqrt in place)
    float* hbuf = ws + nPad;          // [N,128] h' = (A@W)*dis
    float* agg  = hbuf + feat;        // [N,128]
    float* x1   = agg + feat;         // [N,128]
    float* x2   = x1 + feat;          // [N,128]
    float* out  = (float*)d_out;      // [N,64]

    const int B = 256;
    const int gN  = (N + B - 1) / B;
    const int gE  = (E + B - 1) / B;
    const int gF4 = (int)((feat / 4 + B - 1) / B);
    const int gEW = (int)(((long long)E * 32 + B - 1) / B);
    const int gNW = (int)(((long long)N * 32 + B - 1) / B);

    const int mTiles = (N + 15) / 16;                   // 6250
    const dim3 gemmGrid128((mTiles + 3) / 4, KDIM / 64);  // cols = 128 -> grid.y = 2
    const dim3 gemmGridOut((mTiles + 3) / 4, NOUT / 64);  // cols = 64  -> grid.y = 1

    // degrees -> dis = rsqrt(deg)
    k_init_deg<<<gN, B, 0, stream>>>(dis, N);
    k_deg_scatter<<<gE, B, 0, stream>>>(dst, dis, E);
    k_rsqrt<<<gN, B, 0, stream>>>(dis, N);

    // ---- layer 1: h' = (x@W1)*dis ; agg = scatter ; x1 = relu(dis*(agg+h')+b1) ----
    k_gemm_wmma<false, true, false><<<gemmGrid128, 128, 0, stream>>>(x, nullptr, W1, nullptr, dis, hbuf, N, KDIM);
    k_zero4<<<gF4, B, 0, stream>>>((float4*)agg, (int)(feat / 4));
    k_scatter<<<gEW, B, 0, stream>>>(src, dst, hbuf, agg, E);
    k_epilogue<<<gNW, B, 0, stream>>>(agg, hbuf, dis, b1, x1, N);

    // ---- layer 2 ----
    k_gemm_wmma<false, true, false><<<gemmGrid128, 128, 0, stream>>>(x1, nullptr, W2, nullptr, dis, hbuf, N, KDIM);
    k_zero4<<<gF4, B, 0, stream>>>((float4*)agg, (int)(feat / 4));
    k_scatter<<<gEW, B, 0, stream>>>(src, dst, hbuf, agg, E);
    k_epilogue<<<gNW, B, 0, stream>>>(agg, hbuf, dis, b2, x2, N);

    // ---- output projection: (x1 + x2) @ Wo + bo ----
    k_gemm_wmma<true, false, true><<<gemmGridOut, 128, 0, stream>>>(x1, x2, Wo, bo, nullptr, out, N, NOUT);
}